// ModelTorch_68186900792058
// MI455X (gfx1250) — compile-verified
//
#include <hip/hip_runtime.h>
#include <hip/hip_bf16.h>
#include <stdint.h>

// ---------------------------------------------------------------------------
// Reference op: page_size==1 branch of SGLang's assign_req_to_token_pool.
//   out_req_to_token = req_to_token; then for each pid in [0,B):
//     out_req_to_token[req_pool_indices[pid], seq_lens[pid]+j] = out_cache_loc[pid*L+j]
//   plus pass-through of out_cache_loc. Outputs are float32 per harness rules.
//
// Strategy (MI455X, 23.3 TB/s HBM, 192 MB L2):
//   Kernel 1: full 80MB int32 -> 80MB f32 streaming copy/convert, implemented
//             as a double-buffered CDNA5 async-tensor pipeline:
//               global_load_async_to_lds_b128  (ASYNCcnt)
//               s_wait_asynccnt {1|0}
//               ds_load_b128 -> v_cvt_f32_i32 x4 -> global_store_b128
//   Kernel 2: 16K-element scatter overlay + out_cache_loc tail (stream-ordered
//             after kernel 1, which provides the required WAW ordering).
// ---------------------------------------------------------------------------

#define TPB       256          // 8 wave32 waves per workgroup
#define VEC       4            // int32s per lane per tile (16B = B128)
#define TILE_I32  (TPB * VEC)  // 1024 ints = 4 KB per tile
#define POOL_LEN  40960L       // req_to_token row stride (fixed by reference)

__device__ __forceinline__ void async_load_b128(uint32_t lds_byte_addr,
                                                uint64_t sgpr_base,
                                                uint32_t byte_off) {
    // GVS mode: mem_addr = SADDR(64b) + VADDR(32b per-lane byte offset).
    // Writes 16 bytes into LDS at per-lane address lds_byte_addr.
    // Tracked on ASYNCcnt, not LOADcnt.
    asm volatile("global_load_async_to_lds_b128 %0, %1, %2"
                 :
                 : "v"(lds_byte_addr), "v"(byte_off), "s"(sgpr_base)
                 : "memory");
}

__global__ __launch_bounds__(TPB)
void copy_convert_async(const int* __restrict__ src, float* __restrict__ dst,
                        long ntiles) {
    __shared__ int lds[2][TILE_I32];   // 8 KB double buffer (of 320 KB/WGP)

    const int tid = threadIdx.x;
    uint32_t lds_a[2];
    lds_a[0] = (uint32_t)(uintptr_t)&lds[0][tid * VEC];
    lds_a[1] = (uint32_t)(uintptr_t)&lds[1][tid * VEC];
    const uint64_t base = (uint64_t)(uintptr_t)src;

    long t = blockIdx.x;
    const long stride = gridDim.x;
    int buf = 0;

    if (t < ntiles) {
        async_load_b128(lds_a[0], base,
                        (uint32_t)((t * TILE_I32 + tid * VEC) * 4));
    }
    for (; t < ntiles; t += stride) {
        const long tn = t + stride;
        if (tn < ntiles) {
            // Prefetch next tile into the other buffer, then wait until only
            // that prefetch remains outstanding (async loads complete in
            // order, so ASYNCcnt<=1 guarantees the current tile landed).
            async_load_b128(lds_a[buf ^ 1], base,
                            (uint32_t)((tn * TILE_I32 + tid * VEC) * 4));
            asm volatile("s_wait_asynccnt 1" ::: "memory");
        } else {
            asm volatile("s_wait_asynccnt 0" ::: "memory");
        }

        // Each lane consumes exactly the 16B it async-loaded: no barrier
        // needed, just the ASYNCcnt fence above.
        const int4 v = *reinterpret_cast<const int4*>(&lds[buf][tid * VEC]);
        float4 f;
        f.x = (float)v.x;
        f.y = (float)v.y;
        f.z = (float)v.z;
        f.w = (float)v.w;
        *reinterpret_cast<float4*>(dst + t * TILE_I32 + tid * VEC) = f;
        buf ^= 1;
    }
}

// Remainder elements (n not a multiple of TILE_I32); unused at these sizes.
__global__ void copy_convert_tail(const int* __restrict__ src,
                                  float* __restrict__ dst,
                                  long start, long n) {
    long i = start + (long)blockIdx.x * blockDim.x + threadIdx.x;
    if (i < n) dst[i] = (float)src[i];
}

// Scatter overlay + out_cache_loc pass-through. Runs after the bulk copy in
// stream order, so its writes win (matching .at[].set semantics).
__global__ void scatter_assign(const int* __restrict__ req_pool_indices,
                               const int* __restrict__ seq_lens,
                               const int* __restrict__ out_cache_loc,
                               float* __restrict__ out_rtt,
                               float* __restrict__ out_ocl,
                               int L, long pool_len, int total) {
    int idx = blockIdx.x * blockDim.x + threadIdx.x;
    if (idx >= total) return;
    const int pid = idx / L;
    const int j   = idx - pid * L;
    const long row = (long)req_pool_indices[pid];
    const long col = (long)seq_lens[pid] + j;
    const float v = (float)out_cache_loc[idx];
    out_rtt[row * pool_len + col] = v;   // scatter into req_to_token output
    out_ocl[idx] = v;                    // pass-through output segment
}

extern "C" void kernel_launch(void* const* d_in, const int* in_sizes, int n_in,
                              void* d_out, int out_size, void* d_ws, size_t ws_size,
                              hipStream_t stream) {
    // setup_inputs() order:
    // 0 req_pool_indices[B] 1 req_to_token[512*40960] 2 seq_lens[B]
    // 3 extend_lens[B] 4 num_new_pages_per_topk[B] 5 out_cache_loc[B*L]
    // 6 last_page_lens_cumsum[B] 7 duplicate_cache_len 8 topk 9 steps 10 page_size
    const int* req_pool_indices = (const int*)d_in[0];
    const int* req_to_token     = (const int*)d_in[1];
    const int* seq_lens         = (const int*)d_in[2];
    const int* out_cache_loc    = (const int*)d_in[5];
    float* out = (float*)d_out;

    const long n = (long)in_sizes[1];           // 512*40960 elements
    const int  B = in_sizes[0];                 // 256
    const int  L = in_sizes[5] / B;             // topk*steps = 64 (static shape)
    const int  total = B * L;                   // 16384

    // Bulk copy+convert: ~160 MB of traffic, bandwidth-bound (~7us at 23.3 TB/s).
    const long ntiles = n / TILE_I32;           // 20480 full tiles
    if (ntiles > 0) {
        const long want = 4096;                 // grid-stride, ~5 tiles/block
        const int blocks = (int)(ntiles < want ? ntiles : want);
        copy_convert_async<<<blocks, TPB, 0, stream>>>(req_to_token, out, ntiles);
    }
    const long rem_start = ntiles * TILE_I32;
    if (rem_start < n) {
        const long rem = n - rem_start;
        copy_convert_tail<<<(int)((rem + TPB - 1) / TPB), TPB, 0, stream>>>(
            req_to_token, out, rem_start, n);
    }

    // Scatter overlay + out_cache_loc tail (outputs 3/4 are empty).
    scatter_assign<<<(total + TPB - 1) / TPB, TPB, 0, stream>>>(
        req_pool_indices, seq_lens, out_cache_loc,
        out, out + n, L, POOL_LEN, total);
}